// ChamferLoss_78950088835363
// MI455X (gfx1250) — compile-verified
//
#include <hip/hip_runtime.h>
#include <float.h>

typedef __attribute__((ext_vector_type(2))) float v2f;
typedef __attribute__((ext_vector_type(8))) float v8f;

// One direction of the Chamfer loss:
//   for each query point q in qry[b], find min over ref points r of ||q - r||^2,
//   sum those mins per block into partial[blockIdx.x].
//
// Core math per 16(ref) x 16(query) tile, done by ONE v_wmma_f32_16x16x4_f32:
//   A[i, 0..3] = (-2*rx, -2*ry, -2*rz, rr)      (refs on M axis)
//   B[0..3, j] = ( qx,    qy,    qz,   1 )      (queries on N axis)
//   D[i, j]    = rr_i - 2 * dot(r_i, q_j)       (add qq_j once at the very end)
//
// f32 operand layout (ISA 7.12.2): for both the 16x4 A and 4x16 B tiles,
// lane l = (k_pair)*16 + p holds coords (k, k+1) of point p, where
// k = (l/16)*2. So lanes 0-15 carry (x,y) and lanes 16-31 carry (z, slot3).
// D layout: lane l holds column j = l%16 for rows i = r + 8*(l/16), r=0..7.
__global__ __launch_bounds__(256) void chamfer_dir_kernel(
    const float* __restrict__ qry,   // [B, NQ, 3]
    const float* __restrict__ ref,   // [B, NR, 3]
    float* __restrict__ partial,     // [gridDim.x]
    int NQ, int NR, int chunksPerBatch)
{
    const int lane = threadIdx.x & 31;
    const int wave = threadIdx.x >> 5;
    const int b     = blockIdx.x / chunksPerBatch;
    const int chunk = blockIdx.x % chunksPerBatch;

    const int  p  = lane & 15;          // point slot within a 16-tile
    const bool hi = (lane >= 16);       // carries (z, slot3) coord pair

    const size_t qoff = (size_t)b * NQ;
    const size_t roff = (size_t)b * NR;
    const int qbase = chunk * 256 + wave * 32;   // 2 query tiles per wave

    // ---- Load the wave's 2 query tiles as B operands (held for whole scan) ----
    v2f  bmat[2];
    float qq[2];
#pragma unroll
    for (int t = 0; t < 2; ++t) {
        const float* qp = qry + (qoff + (size_t)(qbase + t * 16 + p)) * 3;
        float x = qp[0], y = qp[1], z = qp[2];
        qq[t]     = x * x + y * y + z * z;
        bmat[t].x = hi ? z    : x;
        bmat[t].y = hi ? 1.0f : y;       // K=3 slot of B is the constant 1
    }

    const v8f czero = {0.f, 0.f, 0.f, 0.f, 0.f, 0.f, 0.f, 0.f};
    v8f m0 = {FLT_MAX, FLT_MAX, FLT_MAX, FLT_MAX, FLT_MAX, FLT_MAX, FLT_MAX, FLT_MAX};
    v8f m1 = m0;

    // ---- Scan all refs, 16 at a time ----
    for (int c = 0; c < NR; c += 16) {
        const float* rp = ref + (roff + (size_t)(c + p)) * 3;
        __builtin_prefetch(rp + 192, 0, 0);      // ~4 chunks ahead -> global_prefetch
        float x = rp[0], y = rp[1], z = rp[2];

        v2f a;
        a.x = hi ? (-2.0f * z) : (-2.0f * x);
        a.y = hi ? (x * x + y * y + z * z) : (-2.0f * y);   // K=3 slot of A is rr

        v8f d0 = __builtin_amdgcn_wmma_f32_16x16x4_f32(
            false, a, false, bmat[0], (short)0, czero, false, false);
        v8f d1 = __builtin_amdgcn_wmma_f32_16x16x4_f32(
            false, a, false, bmat[1], (short)0, czero, false, false);

#pragma unroll
        for (int r = 0; r < 8; ++r) {
            m0[r] = fminf(m0[r], d0[r]);
            m1[r] = fminf(m1[r], d1[r]);
        }
    }

    // ---- Fold rows: 8 in-lane mins + cross-half min; then add qq[j] ----
    float acc = 0.0f;
#pragma unroll
    for (int t = 0; t < 2; ++t) {
        v8f& m = t ? m1 : m0;
        float v = m[0];
#pragma unroll
        for (int r = 1; r < 8; ++r) v = fminf(v, m[r]);
        v = fminf(v, __shfl_xor(v, 16, 32));   // combine rows i and i+8
        acc += v + qq[t];                      // deferred ||q||^2
    }

    // Wave sum (each query value duplicated across the two lane halves -> *0.5)
#pragma unroll
    for (int off = 16; off > 0; off >>= 1) acc += __shfl_xor(acc, off, 32);
    acc *= 0.5f;

    __shared__ float smem[8];
    if (lane == 0) smem[wave] = acc;
    __syncthreads();
    if (threadIdx.x == 0) {
        float s = 0.0f;
#pragma unroll
        for (int w = 0; w < 8; ++w) s += smem[w];
        partial[blockIdx.x] = s;
    }
}

// Deterministic fixed-order final reduction: out[b] = mean1 + mean2
__global__ void chamfer_finalize_kernel(const float* __restrict__ p1,
                                        const float* __restrict__ p2,
                                        float* __restrict__ out,
                                        int cpb1, int cpb2,
                                        float invN, float invM, int B)
{
    int b = threadIdx.x;
    if (b < B) {
        float s1 = 0.0f, s2 = 0.0f;
        for (int i = 0; i < cpb1; ++i) s1 += p1[b * cpb1 + i];
        for (int i = 0; i < cpb2; ++i) s2 += p2[b * cpb2 + i];
        out[b] = s1 * invN + s2 * invM;
    }
}

extern "C" void kernel_launch(void* const* d_in, const int* in_sizes, int n_in,
                              void* d_out, int out_size, void* d_ws, size_t ws_size,
                              hipStream_t stream) {
    const int B = 8;
    const float* preds = (const float*)d_in[0];   // [B, N, 3]
    const float* gts   = (const float*)d_in[1];   // [B, M, 3]
    float* out = (float*)d_out;                   // [B]

    const int N = in_sizes[0] / (B * 3);          // 8192
    const int M = in_sizes[1] / (B * 3);          // 8192

    const int cpb1 = N / 256;                     // query chunks per batch, dir 1
    const int cpb2 = M / 256;                     // dir 2
    const int nb1 = B * cpb1;
    const int nb2 = B * cpb2;

    float* partial1 = (float*)d_ws;               // nb1 floats
    float* partial2 = partial1 + nb1;             // nb2 floats

    // loss_1: for each pred (query), min over gts (refs); mean over N preds
    chamfer_dir_kernel<<<nb1, 256, 0, stream>>>(preds, gts, partial1, N, M, cpb1);
    // loss_2: for each gt (query), min over preds (refs); mean over M gts
    chamfer_dir_kernel<<<nb2, 256, 0, stream>>>(gts, preds, partial2, M, N, cpb2);

    chamfer_finalize_kernel<<<1, 32, 0, stream>>>(partial1, partial2, out,
                                                  cpb1, cpb2,
                                                  1.0f / (float)N, 1.0f / (float)M, B);
}